// GNN_4466765987926
// MI455X (gfx1250) — compile-verified
//
#include <hip/hip_runtime.h>

#define N_USER 100000
#define N_ITEM 50000
#define N_NODES_ (N_USER + N_ITEM)
#define EMB 64
#define NNZ_ 4000000
#define BATCH_ 4096
#define NSLOT (3 * BATCH_)   // 12288 gathered rows

typedef __attribute__((ext_vector_type(2))) float v2f;
typedef __attribute__((ext_vector_type(8))) float v8f;

// ---------------------------------------------------------------------------
// Kernel 1: init workspace. map[i] = -1 (row -> compact slot), cprop = 0.
// ---------------------------------------------------------------------------
__global__ void gnn_init_kernel(int* __restrict__ map, float* __restrict__ cprop) {
    int i = blockIdx.x * blockDim.x + threadIdx.x;   // grid covers NSLOT*EMB = 786432
    if (i < N_NODES_) map[i] = -1;
    cprop[i] = 0.0f;                                  // i < NSLOT*EMB by launch config
}

// ---------------------------------------------------------------------------
// Kernel 2: build row -> compact-slot map from the three gather index lists.
// Duplicate rows share whichever slot wins the CAS; sums are identical.
// ---------------------------------------------------------------------------
__global__ void gnn_build_map_kernel(const int* __restrict__ users,
                                     const int* __restrict__ pos,
                                     const int* __restrict__ neg,
                                     int* __restrict__ map) {
    int j = blockIdx.x * blockDim.x + threadIdx.x;
    if (j >= NSLOT) return;
    int row;
    if (j < BATCH_)            row = users[j];
    else if (j < 2 * BATCH_)   row = N_USER + pos[j - BATCH_];
    else                       row = N_USER + neg[j - 2 * BATCH_];
    atomicCAS(&map[row], -1, j);
}

// ---------------------------------------------------------------------------
// Kernel 3: flag-filtered COO SpMM into compact accumulators.
// One thread per nnz for the map test; hot nnz (~8%) are processed
// wave-cooperatively: 32 lanes x float2 = one 64-float embedding row.
// ---------------------------------------------------------------------------
__global__ void gnn_spmm_kernel(const int* __restrict__ rows,
                                const int* __restrict__ cols,
                                const float* __restrict__ vals,
                                const float* __restrict__ ue,
                                const float* __restrict__ ie,
                                const int* __restrict__ map,
                                float* __restrict__ cprop) {
    const int i    = blockIdx.x * blockDim.x + threadIdx.x;   // NNZ_ % 256 == 0
    const int lane = threadIdx.x & 31;

    // stream-ahead prefetch of the row-index stream (global_prefetch_b8)
    __builtin_prefetch(rows + i + 16384, 0, 0);

    const int r = rows[i];
    const int s = map[r];                 // L2-resident 600KB map
    int   c = 0;
    float v = 0.0f;
    if (s >= 0) { c = cols[i]; v = vals[i]; }   // only hot lanes touch cols/vals

    unsigned long long mask = __ballot(s >= 0); // wave32: low 32 bits
    while (mask) {
        const int l = __ffsll(mask) - 1;
        mask &= mask - 1;
        const int   ss = __shfl(s, l, 32);
        const int   cc = __shfl(c, l, 32);
        const float vv = __shfl(v, l, 32);
        const float* src = (cc < N_USER) ? (ue + (size_t)cc * EMB)
                                         : (ie + (size_t)(cc - N_USER) * EMB);
        const float2 x = ((const float2*)src)[lane];
        float* dst = cprop + ss * EMB + lane * 2;
        unsafeAtomicAdd(dst + 0, vv * x.x);     // global_atomic_add_f32, L2 scope
        unsafeAtomicAdd(dst + 1, vv * x.y);
    }
}

// ---------------------------------------------------------------------------
// Kernel 4: combine + gather via WMMA.
// OUT(16x16 tile) = 0.25*E + 0.75*P = sum_k A_k(16x4 scaled-I) x B_k(4x16)
// using V_WMMA_F32_16X16X4_F32. 4 waves/block cover the 4 emb col-tiles.
//
// Layouts (ISA 7.12.2, f32):
//   A 16x4 : lane m=lane&15 holds M=m; VGPR0 = K = 2*(lane>>4), VGPR1 = K+1
//   B 4x16 : lane holds N=lane&15;     VGPR0 = K = 2*(lane>>4), VGPR1 = K+1
//   D 16x16: VGPR r -> M = r + 8*(lane>>4), N = lane&15
// ---------------------------------------------------------------------------
__global__ void gnn_combine_kernel(const int* __restrict__ users,
                                   const int* __restrict__ pos,
                                   const int* __restrict__ neg,
                                   const int* __restrict__ map,
                                   const float* __restrict__ cprop,
                                   const float* __restrict__ ue,
                                   const float* __restrict__ ie,
                                   float* __restrict__ out) {
    const int lane = threadIdx.x & 31;
    const int wave = threadIdx.x >> 5;
    const int m    = lane & 15;     // A-row / B-col / D-col within tile
    const int kh   = lane >> 4;     // lane-half
    const int j0   = blockIdx.x * 16;   // batch-row tile base (768 tiles)
    const int e0   = wave * 16;         // embedding col tile base (4 waves)

    // lane m holds row/slot for batch index j0+m (duplicated across halves)
    const int jj = j0 + m;
    int row;
    if (jj < BATCH_)            row = users[jj];
    else if (jj < 2 * BATCH_)   row = N_USER + pos[jj - BATCH_];
    else                        row = N_USER + neg[jj - 2 * BATCH_];
    const int slot = map[row];          // >= 0: inserted by build_map

    const int col = e0 + m;
    v8f acc = {};

#pragma unroll
    for (int k = 0; k < 4; ++k) {
        const int rA0 = 4 * k + 2 * kh;     // identity column for A-VGPR0
        const int rA1 = rA0 + 1;            // identity column for A-VGPR1
        const int r0 = __shfl(row,  rA0, 32);
        const int r1 = __shfl(row,  rA1, 32);
        const int s0 = __shfl(slot, rA0, 32);
        const int s1 = __shfl(slot, rA1, 32);

        v2f a, b;
        // 0.25 * I x E
        b.x = (r0 < N_USER) ? ue[r0 * EMB + col] : ie[(r0 - N_USER) * EMB + col];
        b.y = (r1 < N_USER) ? ue[r1 * EMB + col] : ie[(r1 - N_USER) * EMB + col];
        a.x = (m == rA0) ? 0.25f : 0.0f;
        a.y = (m == rA1) ? 0.25f : 0.0f;
        acc = __builtin_amdgcn_wmma_f32_16x16x4_f32(false, a, false, b,
                                                    (short)0, acc, false, false);
        // 0.75 * I x P
        b.x = cprop[s0 * EMB + col];
        b.y = cprop[s1 * EMB + col];
        a.x = (m == rA0) ? 0.75f : 0.0f;
        a.y = (m == rA1) ? 0.75f : 0.0f;
        acc = __builtin_amdgcn_wmma_f32_16x16x4_f32(false, a, false, b,
                                                    (short)0, acc, false, false);
    }

#pragma unroll
    for (int r = 0; r < 8; ++r) {
        out[(size_t)(j0 + r + 8 * kh) * EMB + e0 + m] = acc[r];
    }
}

// ---------------------------------------------------------------------------
extern "C" void kernel_launch(void* const* d_in, const int* in_sizes, int n_in,
                              void* d_out, int out_size, void* d_ws, size_t ws_size,
                              hipStream_t stream) {
    const int*   adj_rows = (const int*)  d_in[0];
    const int*   adj_cols = (const int*)  d_in[1];
    const float* adj_vals = (const float*)d_in[2];
    const float* user_emb = (const float*)d_in[3];
    const float* item_emb = (const float*)d_in[4];
    const int*   users    = (const int*)  d_in[5];
    const int*   pos      = (const int*)  d_in[6];
    const int*   neg      = (const int*)  d_in[7];
    float* out = (float*)d_out;

    // workspace: map[N_NODES] (int) | cprop[NSLOT*EMB] (float)  ~= 3.8 MB
    char* ws = (char*)d_ws;
    int*   map   = (int*)ws;
    size_t mapBytes = ((size_t)N_NODES_ * sizeof(int) + 255) & ~(size_t)255;
    float* cprop = (float*)(ws + mapBytes);

    // 1) init: NSLOT*EMB = 786432 elements (covers N_NODES_ = 150000 for map)
    gnn_init_kernel<<<(NSLOT * EMB) / 256, 256, 0, stream>>>(map, cprop);
    // 2) row -> compact slot map
    gnn_build_map_kernel<<<(NSLOT + 255) / 256, 256, 0, stream>>>(users, pos, neg, map);
    // 3) filtered SpMM (NNZ_ = 4e6, divisible by 256)
    gnn_spmm_kernel<<<NNZ_ / 256, 256, 0, stream>>>(adj_rows, adj_cols, adj_vals,
                                                    user_emb, item_emb, map, cprop);
    // 4) WMMA combine + gather: 768 row-tiles x (4 waves = 4 col-tiles)
    gnn_combine_kernel<<<NSLOT / 16, 128, 0, stream>>>(users, pos, neg, map, cprop,
                                                       user_emb, item_emb, out);
}